// FLBROIAlign_20555713479305
// MI455X (gfx1250) — compile-verified
//
#include <hip/hip_runtime.h>

typedef __attribute__((ext_vector_type(2))) float v2f;
typedef __attribute__((ext_vector_type(8))) float v8f;
// 4-byte-aligned 2-float vector: lets clang emit global_load_b64 at DWORD alignment
typedef float v2fu __attribute__((ext_vector_type(2), aligned(4)));

#define NCH 256   // channels
#define NPH 7     // pooled H/W

// One workgroup per ROI. 256 threads = 8 waves (wave32). Each wave handles two
// 16-channel groups. Separable bilinear ROIAlign as two chained f32 WMMA GEMMs:
//   GEMM1: T[ph, j, c] = sum_i WY[ph,i] * feat[c, Y[i], X[j]]   (K=28, 7 chunks of 4)
//   GEMM2: out[ph, pw, c] = sum_j WXT[pw,j] * T[ph, j, c]       (K=28, 7 chunks of 4)
// WY / WXT are block-diagonal (bin = slot>>2) with validity and the 0.25
// averaging factor folded in (0.5 per axis). GEMM2 consumes GEMM1's D registers
// via 2 cross-half shuffles per WMMA: no LDS staging, no barriers.
// The two x-corners of each sample are adjacent pixels -> one b64 gather feeds
// both j-parities (halves VMEM request count, the true bottleneck).
__global__ __launch_bounds__(256) void roi_align_wmma_kernel(
    const float* __restrict__ f0, const float* __restrict__ f1,
    const float* __restrict__ f2, const float* __restrict__ f3,
    const float* __restrict__ rois, const int* __restrict__ level,
    float* __restrict__ out)
{
    const int roi  = blockIdx.x;
    const int tid  = threadIdx.x;
    const int lane = tid & 31;
    const int wave = tid >> 5;
    const int m    = lane & 15;      // matrix row / channel-within-group
    const bool hi  = (lane >= 16);   // upper half-wave

    const int lvl = level[roi];
    const float* feat = (lvl == 0) ? f0 : (lvl == 1) ? f1 : (lvl == 2) ? f2 : f3;
    const int H = 200 >> lvl;                 // 200, 100, 50, 25
    const int W = H;
    const float scale = 0.25f / (float)(1 << lvl);

    const int   bimg = (int)rois[roi * 5 + 0];
    const float x1   = rois[roi * 5 + 1] * scale;
    const float y1   = rois[roi * 5 + 2] * scale;
    const float x2   = rois[roi * 5 + 3] * scale;
    const float y2   = rois[roi * 5 + 4] * scale;
    const float bw = fmaxf(x2 - x1, 1.0f) * (1.0f / 7.0f);  // bin size x
    const float bh = fmaxf(y2 - y1, 1.0f) * (1.0f / 7.0f);  // bin size y

    // ---- per-lane y samples: k-chunk kc uses this lane's sample sy = 2*kc + hi.
    // rowA/rowB = pixel-row offsets of the two y-corners (always in-bounds).
    // A1 chunk kc is nonzero only in matrix row kc -> lane m==kc supplies a1v.
    int rowA[7], rowB[7];
    v2f a1v = {0.f, 0.f};
#pragma unroll
    for (int kc = 0; kc < 7; ++kc) {
        int   sy = 2 * kc + (hi ? 1 : 0);
        float y  = y1 + (float)(sy >> 1) * bh + ((float)(sy & 1) + 0.5f) * (bh * 0.5f);
        float v  = (y > -1.0f && y < (float)H) ? 1.0f : 0.0f;
        float yc = fmaxf(y, 0.0f);
        int   yl = min((int)yc, H - 1);
        int   yh = min(yl + 1, H - 1);
        float ly = (yl >= H - 1) ? 0.0f : (yc - (float)yl);
        float hy = 1.0f - ly;
        rowA[kc] = yl * W;
        rowB[kc] = yh * W;
        if (kc == m) { a1v.x = 0.5f * hy * v; a1v.y = 0.5f * ly * v; }
    }

    // ---- per-lane x sample feeding A2 (chunk g uses lane m==g): sx = 2*m + hi.
    v2f a2v = {0.f, 0.f};
    {
        int   sx = 2 * m + (hi ? 1 : 0);   // only meaningful for m < 7
        float x  = x1 + (float)(sx >> 1) * bw + ((float)(sx & 1) + 0.5f) * (bw * 0.5f);
        float v  = (x > -1.0f && x < (float)W) ? 1.0f : 0.0f;
        float xc = fmaxf(x, 0.0f);
        int   xl = min((int)xc, W - 1);
        float lx = (xl >= W - 1) ? 0.0f : (xc - (float)xl);
        float hx = 1.0f - lx;
        a2v.x = 0.5f * hx * v;
        a2v.y = 0.5f * lx * v;
    }

    const v2f z2 = {0.f, 0.f};
    const v8f z8 = {0.f, 0.f, 0.f, 0.f, 0.f, 0.f, 0.f, 0.f};

#pragma unroll 1
    for (int rep = 0; rep < 2; ++rep) {
        const int grp  = wave + 8 * rep;          // channel group 0..15
        const int chan = grp * 16 + m;            // this lane's channel (column)
        const float* cptr = feat + ((size_t)bimg * NCH + chan) * (size_t)(H * W);

        v8f acc[7];
#pragma unroll
        for (int p = 0; p < 7; ++p) acc[p] = z8;

#pragma unroll 1
        for (int g = 0; g < 7; ++g) {             // j-chunk = GEMM2 K-chunk
            v8f d[4];
#pragma unroll
            for (int half = 0; half < 2; ++half) {
                // x sample sx = 2*g + half covers j slots (4g+2*half, 4g+2*half+1)
                const int sx = 2 * g + half;
                float x  = x1 + (float)(sx >> 1) * bw + ((float)(sx & 1) + 0.5f) * (bw * 0.5f);
                float xc = fmaxf(x, 0.0f);
                int   xl = min((int)xc, W - 1);        // wave-uniform
                int   xb = min(xl, W - 2);             // b64 base, never OOB
                const bool cl = (xl > xb);             // clamped at right edge

                // one b64 gather per (y-corner, k-chunk) feeds both j-parities
                v2fu pa[7], pb[7];
#pragma unroll
                for (int kc = 0; kc < 7; ++kc) {
                    pa[kc] = *(const v2fu*)(cptr + (rowA[kc] + xb));
                    pb[kc] = *(const v2fu*)(cptr + (rowB[kc] + xb));
                }

                // GEMM1: two independent accumulator chains (j even / j odd)
                v8f t0 = z8, t1 = z8;
#pragma unroll
                for (int kc = 0; kc < 7; ++kc) {
                    v2f a = (m == kc) ? a1v : z2;
                    // pixel xl (even slot): .y when clamped, else .x ; pixel xh: .y
                    v2f be = { cl ? pa[kc].y : pa[kc].x, cl ? pb[kc].y : pb[kc].x };
                    v2f bo = { pa[kc].y, pb[kc].y };
                    t0 = __builtin_amdgcn_wmma_f32_16x16x4_f32(
                            false, a, false, be, (short)0, t0, false, false);
                    t1 = __builtin_amdgcn_wmma_f32_16x16x4_f32(
                            false, a, false, bo, (short)0, t1, false, false);
                }
                d[2 * half]     = t0;
                d[2 * half + 1] = t1;
            }

            // GEMM2: acc[ph] += WXT(chunk g) x T(chunk g). B2 rows 2,3 live in
            // lanes 0..15 of d[2]/d[3] -> shuffle across the half-wave split.
#pragma unroll
            for (int p = 0; p < 7; ++p) {
                float s2 = __shfl(d[2][p], m, 32);
                float s3 = __shfl(d[3][p], m, 32);
                v2f b2;
                b2.x = hi ? s2 : d[0][p];
                b2.y = hi ? s3 : d[1][p];
                v2f a2 = (m == g) ? a2v : z2;
                acc[p] = __builtin_amdgcn_wmma_f32_16x16x4_f32(
                            false, a2, false, b2, (short)0, acc[p], false, false);
            }
        }

        // D rows = pw: valid rows 0..6 live in lanes 0..15
        if (!hi) {
            const size_t obase = ((size_t)roi * NCH + chan) * (size_t)(NPH * NPH);
#pragma unroll
            for (int p = 0; p < 7; ++p) {
#pragma unroll
                for (int r = 0; r < 7; ++r) {
                    out[obase + (size_t)(p * 7 + r)] = acc[p][r];
                }
            }
        }
    }
}

extern "C" void kernel_launch(void* const* d_in, const int* in_sizes, int n_in,
                              void* d_out, int out_size, void* d_ws, size_t ws_size,
                              hipStream_t stream) {
    const float* f0   = (const float*)d_in[0];
    const float* f1   = (const float*)d_in[1];
    const float* f2   = (const float*)d_in[2];
    const float* f3   = (const float*)d_in[3];
    const float* rois = (const float*)d_in[4];
    // d_in[5] = rois_counts (unused: rois carry their batch index)
    const int*   level = (const int*)d_in[6];
    float* out = (float*)d_out;

    const int K = in_sizes[4] / 5;   // number of ROIs
    roi_align_wmma_kernel<<<K, 256, 0, stream>>>(f0, f1, f2, f3, rois, level, out);
}